// ConvTemporalGraphical_35115652612184
// MI455X (gfx1250) — compile-verified
//
#include <hip/hip_runtime.h>

typedef __attribute__((ext_vector_type(2))) float v2f;
typedef __attribute__((ext_vector_type(4))) float v4f;
typedef __attribute__((ext_vector_type(8))) float v8f;
typedef __attribute__((ext_vector_type(4))) int   v4i;

typedef __attribute__((address_space(1))) v4i* gptr_v4i;
typedef __attribute__((address_space(3))) v4i* lptr_v4i;

#define N_ 64
#define C_ 64
#define T_ 512
#define V_ 25
#define O_ 8
#define K_ 3

#define BP_ELEMS (O_ * 28 * 32)        // 7168 floats, zero-padded B
#define ROWS_PER_BLOCK 64              // 4 row-tiles of 16 per wave

#if defined(__gfx1250__) && \
    __has_builtin(__builtin_amdgcn_global_load_async_to_lds_b128) && \
    __has_builtin(__builtin_amdgcn_s_wait_asynccnt)
#define USE_ASYNC_LDS 1
#else
#define USE_ASYNC_LDS 0
#endif

// --- one-shot prep: Bp[o][v:28][w:32] = sum_k A[k,v,w]*imp[o,k,v,w] (padded) ---
__global__ void __launch_bounds__(256)
prep_kernel(const float* __restrict__ A, const float* __restrict__ imp,
            float* __restrict__ Bp) {
    int idx = blockIdx.x * blockDim.x + threadIdx.x;
    if (idx < BP_ELEMS) {
        const int o   = idx / (28 * 32);
        const int rem = idx % (28 * 32);
        const int v   = rem >> 5;
        const int w   = rem & 31;
        float s = 0.0f;
        if (v < V_ && w < V_) {
            #pragma unroll
            for (int k = 0; k < K_; ++k)
                s += A[(k * V_ + v) * V_ + w] *
                     imp[((o * K_ + k) * V_ + v) * V_ + w];
        }
        Bp[idx] = s;
    }
}

// --- tiny elementwise kernel: A_full[o,k,v,w] = A[k,v,w] * importance[o,k,v,w] ---
__global__ void __launch_bounds__(256)
afull_kernel(const float* __restrict__ A, const float* __restrict__ imp,
             float* __restrict__ out) {
    int idx = blockIdx.x * blockDim.x + threadIdx.x;
    if (idx < O_ * K_ * V_ * V_) {
        out[idx] = A[idx % (K_ * V_ * V_)] * imp[idx];
    }
}

// --- main GEMM: y[n,o,t,w] = sum_v x[n,0,t,v] * B[o,v,w] ---
// Rows = (n,t) pairs (32768). Block = 256 threads = 8 wave32s; wave i owns o=i.
// Block handles 64 rows (4 row-tiles of 16 per wave); each row-tile produces
// two 16-wide column tiles via 7x V_WMMA_F32_16X16X4_F32 (v padded 25 -> 28).
__global__ void __launch_bounds__(256)
gemm_kernel(const float* __restrict__ x, const float* __restrict__ Bp,
            float* __restrict__ y) {
    __shared__ float Bs[BP_ELEMS];     // 28 KB

    const int tid = threadIdx.x;

    // Copy precomputed padded B into LDS (1792 x b128 = 7 per thread).
#if USE_ASYNC_LDS
    {
        const v4f* g4 = (const v4f*)Bp;
        v4f*       l4 = (v4f*)Bs;
        #pragma unroll
        for (int i = 0; i < 7; ++i) {
            const int idx = tid + i * 256;
            __builtin_amdgcn_global_load_async_to_lds_b128(
                (gptr_v4i)(g4 + idx),
                (lptr_v4i)(l4 + idx),
                0, 0);
        }
        __builtin_amdgcn_s_wait_asynccnt(0);
    }
#else
    {
        const v4f* g4 = (const v4f*)Bp;
        v4f*       l4 = (v4f*)Bs;
        #pragma unroll
        for (int i = 0; i < 7; ++i) {
            const int idx = tid + i * 256;
            l4[idx] = g4[idx];
        }
    }
#endif
    __syncthreads();

    const int lane = tid & 31;
    const int o    = tid >> 5;         // wave index == output group o
    const int m    = lane & 15;        // row (A) / col (B,D) within 16x16 tile
    const int h    = lane >> 4;        // half-wave selector per ISA layout
    const float* Bo = Bs + o * 28 * 32;

    for (int sub = 0; sub < 4; ++sub) {
        const int rowbase = (blockIdx.x << 6) + (sub << 4);

        // A fragment: x[n,0,t,v]; flat x offset = (n*C_*T_ + t)*V_
        const int row = rowbase + m;
        const int n   = row >> 9;
        const int t   = row & (T_ - 1);
        const float* xr = x + ((size_t)n * C_ * T_ + (size_t)t) * V_;

        v2f afrag[7];
        #pragma unroll
        for (int kk = 0; kk < 7; ++kk) {
            const int v0 = kk * 4 + 2 * h;     // VGPR0 holds K=2h, VGPR1 K=2h+1
            afrag[kk].x = (v0 < V_)     ? xr[v0]     : 0.0f;
            afrag[kk].y = (v0 + 1 < V_) ? xr[v0 + 1] : 0.0f;
        }

        #pragma unroll
        for (int ct = 0; ct < 2; ++ct) {
            v8f acc = {0.f, 0.f, 0.f, 0.f, 0.f, 0.f, 0.f, 0.f};
            #pragma unroll
            for (int kk = 0; kk < 7; ++kk) {
                const int v0 = kk * 4 + 2 * h;
                v2f b;
                b.x = Bo[v0 * 32 + ct * 16 + m];        // B[v0,   col]
                b.y = Bo[(v0 + 1) * 32 + ct * 16 + m];  // B[v0+1, col]
                acc = __builtin_amdgcn_wmma_f32_16x16x4_f32(
                    /*neg_a=*/false, afrag[kk], /*neg_b=*/false, b,
                    /*c_mod=*/(short)0, acc, /*reuse_a=*/false, /*reuse_b=*/false);
            }
            const int c = ct * 16 + m;                  // output w
            if (c < V_) {
                #pragma unroll
                for (int r = 0; r < 8; ++r) {           // D VGPR r -> row r + 8h
                    const int rm = rowbase + r + 8 * h;
                    const int nn = rm >> 9;
                    const int tt = rm & (T_ - 1);
                    y[((size_t)(nn * O_ + o) * T_ + (size_t)tt) * V_ + c] = acc[r];
                }
            }
        }
    }
}

extern "C" void kernel_launch(void* const* d_in, const int* in_sizes, int n_in,
                              void* d_out, int out_size, void* d_ws, size_t ws_size,
                              hipStream_t stream) {
    (void)in_sizes; (void)n_in; (void)out_size; (void)ws_size;
    const float* x   = (const float*)d_in[0];  // (64,64,512,25)
    const float* A   = (const float*)d_in[1];  // (3,25,25)
    const float* imp = (const float*)d_in[2];  // (8,3,25,25)

    float* y     = (float*)d_out;                         // (64,8,512,25)
    float* afull = y + (size_t)N_ * O_ * T_ * V_;         // (8,3,25,25)
    float* Bp    = (float*)d_ws;                          // 7168 floats scratch

    // 1) Fold k and importance into padded B (one tiny pass).
    prep_kernel<<<(BP_ELEMS + 255) / 256, 256, 0, stream>>>(A, imp, Bp);

    // 2) GEMM over all (n,t) rows: 32768 / 64 = 512 blocks of 256 threads.
    gemm_kernel<<<(N_ * T_) / ROWS_PER_BLOCK, 256, 0, stream>>>(x, Bp, y);

    // 3) A_full elementwise (15000 elements).
    afull_kernel<<<(O_ * K_ * V_ * V_ + 255) / 256, 256, 0, stream>>>(A, imp, afull);
}